// GridSubsample_74998718923128
// MI455X (gfx1250) — compile-verified
//
#include <hip/hip_runtime.h>
#include <stdint.h>

// ---------------- configuration ----------------
#define VOXF 0.04f
#define PER_BATCH_BINS (101*101*101)   // 1030301 bins per cloud (<=101 cells/axis)
#define BIN_TILE 256                   // points per binning block
#define C_BLOCK 256
#define C_ITEMS 8
#define C_TILE (C_BLOCK*C_ITEMS)       // 2048 bins per compaction block
#define MAX_CBLKS 8192

// header byte offsets inside workspace
#define H_MINBITS 0       // 3 x u32 (float bits of per-axis min; data is nonnegative)
#define H_MAXBITS 16      // 3 x u32
#define H_DIMS    32      // 3 x i32
#define H_STRIDES 48      // sB, sX, sY
#define H_CURSOR  64      // running output-row cursor (u32)
#define H_NEWLEN  128     // B x i32
#define H_OFFSETS 4096    // (B+1) x i32 batch prefix
#define H_BLKCNT  8192    // u32[MAX_CBLKS]
#define H_BLKOFF  (H_BLKCNT + MAX_CBLKS*4)
#define BIN_BASE  (H_BLKOFF + MAX_CBLKS*4)   // 73728, 256B aligned

// gfx1250 async global->LDS path (guarded; plain-load fallback keeps compile clean)
#if defined(__has_builtin)
#  if __has_builtin(__builtin_amdgcn_global_load_async_to_lds_b128) && \
      __has_builtin(__builtin_amdgcn_s_wait_asynccnt)
#    define USE_ASYNC_LDS 1
#  endif
#endif
#ifndef USE_ASYNC_LDS
#  define USE_ASYNC_LDS 0
#endif

#if USE_ASYNC_LDS
typedef int v4i __attribute__((ext_vector_type(4)));
typedef __attribute__((address_space(1))) v4i* gptr_b128;
typedef __attribute__((address_space(3))) v4i* lptr_b128;
#endif

// ---------------- kernels ----------------

__global__ void k_init(uint32_t* hdr, const int* __restrict__ lengths, int B) {
  const int t = threadIdx.x;
  if (t < 3) { hdr[H_MINBITS/4 + t] = 0x7f7fffffu; hdr[H_MAXBITS/4 + t] = 0u; }
  if (t == 0) {
    hdr[H_CURSOR/4] = 0u;
    int* offs = (int*)hdr + H_OFFSETS/4;
    int acc = 0; offs[0] = 0;
    for (int b = 0; b < B; ++b) { acc += lengths[b]; offs[b+1] = acc; }
  }
  int* nl = (int*)hdr + H_NEWLEN/4;
  for (int i = t; i < B; i += blockDim.x) nl[i] = 0;
}

__global__ __launch_bounds__(256) void k_minmax(const float* __restrict__ pts,
                                                long long N, uint32_t* hdr) {
  __shared__ float smn[3][256];
  __shared__ float smx[3][256];
  const int t = threadIdx.x;
  float mn[3] = {3.4e38f, 3.4e38f, 3.4e38f};
  float mx[3] = {0.f, 0.f, 0.f};   // data is nonnegative
  const long long stride = (long long)gridDim.x * blockDim.x;
  for (long long i = (long long)blockIdx.x * blockDim.x + t; i < N; i += stride) {
    __builtin_prefetch(pts + (i + stride) * 3, 0, 0);   // global_prefetch_b8
    // non-temporal: this pass must not evict the (soon hot) L2 bin arrays
    float a = __builtin_nontemporal_load(&pts[i*3+0]);
    float b = __builtin_nontemporal_load(&pts[i*3+1]);
    float c = __builtin_nontemporal_load(&pts[i*3+2]);
    mn[0] = fminf(mn[0], a); mx[0] = fmaxf(mx[0], a);
    mn[1] = fminf(mn[1], b); mx[1] = fmaxf(mx[1], b);
    mn[2] = fminf(mn[2], c); mx[2] = fmaxf(mx[2], c);
  }
  for (int c = 0; c < 3; ++c) { smn[c][t] = mn[c]; smx[c][t] = mx[c]; }
  __syncthreads();
  for (int s = 128; s > 0; s >>= 1) {
    if (t < s) {
      for (int c = 0; c < 3; ++c) {
        smn[c][t] = fminf(smn[c][t], smn[c][t+s]);
        smx[c][t] = fmaxf(smx[c][t], smx[c][t+s]);
      }
    }
    __syncthreads();
  }
  if (t == 0) {
    for (int c = 0; c < 3; ++c) {
      atomicMin((int*)&hdr[H_MINBITS/4 + c], __float_as_int(smn[c][0]));
      atomicMax((int*)&hdr[H_MAXBITS/4 + c], __float_as_int(smx[c][0]));
    }
  }
}

__global__ void k_dims(uint32_t* hdr) {
  int* hi = (int*)hdr;
  int d[3];
  for (int c = 0; c < 3; ++c) {
    float mn = __uint_as_float(hdr[H_MINBITS/4 + c]);
    float mx = __uint_as_float(hdr[H_MAXBITS/4 + c]);
    int dd = (int)floorf((mx - mn) / VOXF) + 1;
    if (dd < 1) dd = 1;
    if (dd > 101) dd = 101;
    d[c] = dd; hi[H_DIMS/4 + c] = dd;
  }
  hi[H_STRIDES/4 + 2] = d[2];
  hi[H_STRIDES/4 + 1] = d[1] * d[2];
  hi[H_STRIDES/4 + 0] = d[0] * d[1] * d[2];
}

// vectorized clear for the bin arrays (regular temporal: re-read immediately)
__global__ void k_zero4(uint4* __restrict__ p, long long nvec) {
  long long i = (long long)blockIdx.x * blockDim.x + threadIdx.x;
  const long long stride = (long long)gridDim.x * blockDim.x;
  uint4 z; z.x = 0u; z.y = 0u; z.z = 0u; z.w = 0u;
  for (; i < nvec; i += stride) p[i] = z;
}

// zero only pooled rows >= n_voxels (device bound from cursor), non-temporal
__global__ void k_zero_tail(float* __restrict__ out, const uint32_t* __restrict__ cursor,
                            long long rows) {
  const long long start = (long long)(*cursor) * 3;
  const long long n = rows * 3;
  long long i = start + (long long)blockIdx.x * blockDim.x + threadIdx.x;
  const long long stride = (long long)gridDim.x * blockDim.x;
  for (; i < n; i += stride) __builtin_nontemporal_store(0.0f, &out[i]);
}

__global__ __launch_bounds__(256) void k_bin(const float* __restrict__ pts, long long N,
    const uint32_t* __restrict__ hdr, const int* __restrict__ goffs,
    uint32_t* __restrict__ cnt, float* __restrict__ sx, float* __restrict__ sy,
    float* __restrict__ sz, int b0, int b1, int B)
{
  __shared__ int offs[257];
#if USE_ASYNC_LDS
  __shared__ __attribute__((aligned(16))) float lp[3 * BIN_TILE];
#endif
  const int t = threadIdx.x;
  const int nb = (B < 256) ? B : 256;
  for (int i = t; i <= nb; i += 256) offs[i] = goffs[i];
  const int* hi = (const int*)hdr;
  const int dx = hi[H_DIMS/4+0], dy = hi[H_DIMS/4+1], dz = hi[H_DIMS/4+2];
  const int sB = hi[H_STRIDES/4+0], sX = hi[H_STRIDES/4+1], sY = hi[H_STRIDES/4+2];
  const float mn0 = __uint_as_float(hdr[H_MINBITS/4+0]);
  const float mn1 = __uint_as_float(hdr[H_MINBITS/4+1]);
  const float mn2 = __uint_as_float(hdr[H_MINBITS/4+2]);
  const long long base = (long long)blockIdx.x * BIN_TILE;
  const bool full = (base + BIN_TILE) <= N;
  __syncthreads();

  float x = 0.f, y = 0.f, z = 0.f;
  bool have = false;
#if USE_ASYNC_LDS
  if (full) {
    const float* gsrc = pts + base * 3;
    if (t < (3 * BIN_TILE * 4) / 16) {  // 192 x 16B = full tile
      __builtin_amdgcn_global_load_async_to_lds_b128(
          (gptr_b128)gsrc + t,
          (lptr_b128)lp + t, 0, 0);
    }
    __builtin_amdgcn_s_wait_asynccnt(0);
    __syncthreads();
    x = lp[3*t+0]; y = lp[3*t+1]; z = lp[3*t+2];
    have = true;
  }
#endif
  const long long i = base + t;
  if (!have) {
    if (i < N) { x = pts[i*3+0]; y = pts[i*3+1]; z = pts[i*3+2]; }
  }
  if (i >= N) return;

  // batch id via binary search over prefix offsets
  int lo = 0, hi2 = B;
  while (hi2 - lo > 1) {
    int mid = (lo + hi2) >> 1;
    int ov = (mid <= nb) ? offs[mid] : goffs[mid];
    if (i >= (long long)ov) lo = mid; else hi2 = mid;
  }
  if (lo < b0 || lo >= b1) return;

  int cx = (int)floorf((x - mn0) / VOXF);
  int cy = (int)floorf((y - mn1) / VOXF);
  int cz = (int)floorf((z - mn2) / VOXF);
  cx = cx < 0 ? 0 : (cx > dx-1 ? dx-1 : cx);
  cy = cy < 0 ? 0 : (cy > dy-1 ? dy-1 : cy);
  cz = cz < 0 ? 0 : (cz > dz-1 ? dz-1 : cz);
  const int bin = (lo - b0) * sB + cx * sX + cy * sY + cz;
  atomicAdd(&cnt[bin], 1u);
  atomicAdd(&sx[bin], x);
  atomicAdd(&sy[bin], y);
  atomicAdd(&sz[bin], z);
}

__global__ __launch_bounds__(256) void k_count(const uint32_t* __restrict__ cnt, int nbins,
    uint32_t* __restrict__ blkCnt, const uint32_t* __restrict__ hdr,
    int* __restrict__ newLen, int b0, int bpp)
{
  __shared__ uint32_t part[256];
  __shared__ int bc[128];
  const int t = threadIdx.x;
  for (int i = t; i < 128; i += 256) bc[i] = 0;
  __syncthreads();
  const int sB = ((const int*)hdr)[H_STRIDES/4 + 0];
  const int base = blockIdx.x * C_TILE + t * C_ITEMS;
  uint32_t c = 0;
  for (int k = 0; k < C_ITEMS; ++k) {
    int bin = base + k;
    if (bin < nbins && cnt[bin] != 0u) {
      c++;
      int g = bin / sB; if (g > 127) g = 127;
      atomicAdd(&bc[g], 1);
    }
  }
  part[t] = c;
  __syncthreads();
  for (int s = 128; s > 0; s >>= 1) {
    if (t < s) part[t] += part[t+s];
    __syncthreads();
  }
  if (t == 0) blkCnt[blockIdx.x] = part[0];
  for (int i = t; i < bpp && i < 128; i += 256)
    if (bc[i]) atomicAdd(&newLen[b0 + i], bc[i]);
}

__global__ __launch_bounds__(1024) void k_scan(const uint32_t* __restrict__ blkCnt,
    uint32_t* __restrict__ blkOff, int n, uint32_t* __restrict__ cursor)
{
  __shared__ uint32_t tsum[1024];
  __shared__ uint32_t curS;
  const int t = threadIdx.x;
  if (t == 0) curS = *cursor;
  const int per = (n + 1023) / 1024;
  const int lo = t * per;
  const int hiE = (lo + per < n) ? lo + per : n;
  uint32_t s = 0;
  for (int i = lo; i < hiE; ++i) s += blkCnt[i];
  tsum[t] = s;
  __syncthreads();
  for (int off = 1; off < 1024; off <<= 1) {
    uint32_t v = (t >= off) ? tsum[t - off] : 0u;
    __syncthreads();
    tsum[t] += v;
    __syncthreads();
  }
  uint32_t run = curS + (tsum[t] - s);
  for (int i = lo; i < hiE; ++i) { blkOff[i] = run; run += blkCnt[i]; }
  if (t == 0) *cursor = curS + tsum[1023];
}

__global__ __launch_bounds__(256) void k_emit(const uint32_t* __restrict__ cnt,
    const float* __restrict__ sx, const float* __restrict__ sy,
    const float* __restrict__ sz, int nbins,
    const uint32_t* __restrict__ blkOff, float* __restrict__ out)
{
  __shared__ uint32_t psum[256];
  const int t = threadIdx.x;
  const int base = blockIdx.x * C_TILE + t * C_ITEMS;
  uint32_t cc[C_ITEMS];
  uint32_t c = 0;
  for (int k = 0; k < C_ITEMS; ++k) {
    int bin = base + k;
    cc[k] = (bin < nbins) ? cnt[bin] : 0u;
    c += (cc[k] != 0u) ? 1u : 0u;
  }
  psum[t] = c;
  __syncthreads();
  for (int off = 1; off < 256; off <<= 1) {
    uint32_t v = (t >= off) ? psum[t - off] : 0u;
    __syncthreads();
    psum[t] += v;
    __syncthreads();
  }
  uint32_t pos = blkOff[blockIdx.x] + (psum[t] - c);
  for (int k = 0; k < C_ITEMS; ++k) {
    if (cc[k]) {
      int bin = base + k;
      float inv = 1.0f / (float)cc[k];
      // non-temporal: output is write-once, keep it out of L2
      __builtin_nontemporal_store(sx[bin] * inv, &out[(size_t)pos*3 + 0]);
      __builtin_nontemporal_store(sy[bin] * inv, &out[(size_t)pos*3 + 1]);
      __builtin_nontemporal_store(sz[bin] * inv, &out[(size_t)pos*3 + 2]);
      pos++;
    }
  }
}

__global__ void k_final(const int* __restrict__ newLen, float* __restrict__ outTail, int B) {
  int t = blockIdx.x * blockDim.x + threadIdx.x;
  if (t < B) outTail[t] = (float)newLen[t];
}

// ---------------- host launcher ----------------

extern "C" void kernel_launch(void* const* d_in, const int* in_sizes, int n_in,
                              void* d_out, int out_size, void* d_ws, size_t ws_size,
                              hipStream_t stream) {
  (void)n_in;
  const float* pts = (const float*)d_in[0];
  const int* lengths = (const int*)d_in[1];
  const long long N = (long long)in_sizes[0] / 3;
  const int B = in_sizes[1];

  uint8_t* ws = (uint8_t*)d_ws;
  uint32_t* hdr = (uint32_t*)ws;
  const int* goffs = (const int*)(ws + H_OFFSETS);
  int* newLen = (int*)(ws + H_NEWLEN);
  uint32_t* blkCnt = (uint32_t*)(ws + H_BLKCNT);
  uint32_t* blkOff = (uint32_t*)(ws + H_BLKOFF);
  uint32_t* cursor = (uint32_t*)(ws + H_CURSOR);

  // how many batches' bins fit in workspace at once
  const long long perBatchBytes = (long long)PER_BATCH_BINS * 16;
  long long avail = (ws_size > (size_t)BIN_BASE) ? (long long)ws_size - BIN_BASE : 0;
  int bpp = (int)(avail / perBatchBytes);
  const int bppMaxBlocks = (int)(((long long)MAX_CBLKS * C_TILE) / PER_BATCH_BINS); // 16
  if (bpp > bppMaxBlocks) bpp = bppMaxBlocks;
  if (bpp > B) bpp = B;
  if (bpp < 1) bpp = 1;
  const int phases = (B + bpp - 1) / bpp;
  const long long cap = (long long)bpp * PER_BATCH_BINS;   // bins per phase

  uint32_t* binCnt = (uint32_t*)(ws + BIN_BASE);
  float* sx = (float*)(ws + BIN_BASE + (size_t)cap * 4);
  float* sy = sx + cap;
  float* sz = sy + cap;
  const int nCBlk = (int)((cap + C_TILE - 1) / C_TILE);

  // one-time setup
  k_init<<<1, 256, 0, stream>>>(hdr, lengths, B);
  k_minmax<<<1024, 256, 0, stream>>>(pts, N, hdr);
  k_dims<<<1, 1, 0, stream>>>(hdr);

  const int binTiles = (int)((N + BIN_TILE - 1) / BIN_TILE);
  const long long znvec = cap;               // cnt+sx+sy+sz = 4*cap dwords = cap uint4
  const int zblks = (int)((znvec + 255) / 256);

  for (int p = 0; p < phases; ++p) {
    const int b0 = p * bpp;
    const int b1 = (b0 + bpp < B) ? b0 + bpp : B;
    k_zero4<<<zblks, 256, 0, stream>>>((uint4*)binCnt, znvec);
    k_bin<<<binTiles, 256, 0, stream>>>(pts, N, hdr, goffs, binCnt, sx, sy, sz, b0, b1, B);
    k_count<<<nCBlk, 256, 0, stream>>>(binCnt, (int)cap, blkCnt, hdr, newLen, b0, b1 - b0);
    k_scan<<<1, 1024, 0, stream>>>(blkCnt, blkOff, nCBlk, cursor);
    k_emit<<<nCBlk, 256, 0, stream>>>(binCnt, sx, sy, sz, (int)cap, blkOff, (float*)d_out);
  }

  // zero only the pooled rows beyond n_voxels (bound read on device from cursor)
  {
    long long n3 = N * 3;
    int blks = (int)((n3 + 255) / 256);
    k_zero_tail<<<blks, 256, 0, stream>>>((float*)d_out, cursor, N);
  }

  // new_lengths at the tail of the concatenated output, as float values
  float* outTail = (float*)d_out + (out_size - B);
  k_final<<<(B + 255) / 256, 256, 0, stream>>>(newLen, outTail, B);
}